// sLSTMCell_42829413876127
// MI455X (gfx1250) — compile-verified
//
#include <hip/hip_runtime.h>
#include <cstdint>

namespace {

constexpr int kS = 512;
constexpr int kB = 32;
constexpr int kD = 512;
constexpr int kH = 512;
constexpr int kBH  = kB * kH;       // 16384
constexpr int kSBH = kS * kBH;      // 8388608
constexpr int kM   = kS * kB;       // 16384 rows of the big GEMMs

typedef __attribute__((ext_vector_type(2))) float v2f;
typedef __attribute__((ext_vector_type(8))) float v8f;
typedef __attribute__((ext_vector_type(4))) unsigned int v4u;
typedef __attribute__((ext_vector_type(8))) int v8i;
typedef __attribute__((ext_vector_type(4))) int v4i;

__device__ __forceinline__ float sigm(float x) { return 1.0f / (1.0f + expf(-x)); }

__device__ __forceinline__ v8f zero8() {
  v8f z = {0.f, 0.f, 0.f, 0.f, 0.f, 0.f, 0.f, 0.f};
  return z;
}

// A-frag (16x4 f32, ISA layout): lane<16 -> row=lane, K={k0,k0+1}; lane>=16 -> row=lane-16, K={k0+2,k0+3}
__device__ __forceinline__ v2f load_a(const float* tile, int ld, int k0) {
  int lane = threadIdx.x & 31;
  int m = lane & 15;
  int k = k0 + ((lane >> 4) << 1);
  v2f a;
  a.x = tile[(size_t)m * ld + k];
  a.y = tile[(size_t)m * ld + k + 1];
  return a;
}

// B-frag (4x16 f32) where the GEMM operand is B[k][n] = W[n*ld + k] (W stored row-major over n)
__device__ __forceinline__ v2f load_bt(const float* w, int ld, int k0) {
  int lane = threadIdx.x & 31;
  int n = lane & 15;
  int k = k0 + ((lane >> 4) << 1);
  v2f b;
  b.x = w[(size_t)n * ld + k];
  b.y = w[(size_t)n * ld + k + 1];
  return b;
}

__device__ __forceinline__ v8f wmma4(v2f a, v2f b, v8f c) {
  // v_wmma_f32_16x16x4_f32 : D(16x16,f32) = A(16x4) * B(4x16) + C
  return __builtin_amdgcn_wmma_f32_16x16x4_f32(false, a, false, b, (short)0, c, false, false);
}

// K-chunk / LDS tiling parameters for the staged GEMM kernels
constexpr int KC  = 32;        // k-chunk width (one LDS row per TDM pad interval)
constexpr int KC4 = KC / 4;    // float4s per row
constexpr int LDA = KC + 4;    // 144B row pitch: 16B aligned, bank-conflict free
constexpr int LDB = KC + 4;    // must equal KC + TDM pad_amount (4 DWORDs)

#if __has_builtin(__builtin_amdgcn_tensor_load_to_lds)
#define HAVE_TDM 1
#else
#define HAVE_TDM 0
#endif

// ---------------------------------------------------------------------------
// Stage 1: mask h0/c0 -> h_w1, c_w1
__global__ __launch_bounds__(256) void prep_kernel(const float* __restrict__ h0,
                                                   const float* __restrict__ c0,
                                                   const int* __restrict__ seq_lens,
                                                   float* __restrict__ hw1,
                                                   float* __restrict__ cw1) {
  int idx = blockIdx.x * 256 + threadIdx.x;
  if (idx >= kSBH) return;
  int s = idx / kBH;
  int b = (idx / kH) % kB;
  bool masked = s >= seq_lens[b];
  hw1[idx] = masked ? 0.0f : h0[idx];
  cw1[idx] = masked ? 0.0f : c0[idx];
}

// Stage 2: h_hat = mean over S of h_w1
__global__ __launch_bounds__(256) void hhat_kernel(const float* __restrict__ hw1,
                                                   float* __restrict__ hhat) {
  int idx = blockIdx.x * 256 + threadIdx.x;  // b*H + h
  if (idx >= kBH) return;
  float acc = 0.f;
  for (int s = 0; s < kS; ++s) acc += hw1[(size_t)s * kBH + idx];
  hhat[idx] = acc * (1.0f / (float)kS);
}

// Stage 3: small (32 x 512, K=512) WMMA GEMMs: fg, og, gbias1, vterm[0..6] (vterm folds Wb)
__global__ __launch_bounds__(32) void small_gemm_kernel(
    const float* __restrict__ h0, const float* __restrict__ hhat,
    const float* __restrict__ Sw, const float* __restrict__ Su, const float* __restrict__ Sb,
    const float* __restrict__ Wv, const float* __restrict__ Wb,
    float* __restrict__ fg, float* __restrict__ og,
    float* __restrict__ gb1, float* __restrict__ vterm) {
  const float* hg1 = h0 + (size_t)kS * kBH;  // h0[-1]
  int lane = threadIdx.x & 31;
  int n0 = blockIdx.x * 16;
  int row0 = blockIdx.y * 16;
  int job = blockIdx.z;

  const float* W1;
  const float* X2 = nullptr;
  const float* W2 = nullptr;
  const float* bias;
  float* out;
  bool act = false;
  if (job == 0)      { W1 = Sw;                        X2 = hhat; W2 = Su;                        bias = Sb;           out = fg;  act = true; }
  else if (job == 1) { W1 = Sw + 2 * (size_t)kH * kH;  X2 = hhat; W2 = Su + 2 * (size_t)kH * kH;  bias = Sb + 2 * kH;  out = og;  act = true; }
  else if (job == 2) { W1 = Sw + (size_t)kH * kH;                                                 bias = Sb + kH;      out = gb1; }
  else { int g = job - 3; W1 = Wv + (size_t)g * kH * kH;                                          bias = Wb + g * kH;  out = vterm + (size_t)g * kBH; }

  v8f acc = zero8();
  {
    const float* A = hg1 + (size_t)row0 * kH;
    const float* Wt = W1 + (size_t)n0 * kH;
    for (int k = 0; k < kH; k += 4)
      acc = wmma4(load_a(A, kH, k), load_bt(Wt, kH, k), acc);
  }
  if (X2) {
    const float* A = X2 + (size_t)row0 * kH;
    const float* Wt = W2 + (size_t)n0 * kH;
    for (int k = 0; k < kH; k += 4)
      acc = wmma4(load_a(A, kH, k), load_bt(Wt, kH, k), acc);
  }
  int hcol = n0 + (lane & 15);
#pragma unroll
  for (int e = 0; e < 8; ++e) {
    int m = e + ((lane >> 4) << 3);
    int b = row0 + m;
    float val = acc[e] + bias[hcol];
    if (act) val = sigm(val);
    out[(size_t)b * kH + hcol] = val;
  }
}

// Stage 4: fi = sigmoid(h_w1 @ Su[1]^T + gbias1)   (M = S*B, N = H, K = H)
// LDS-staged: 256 threads = 8 waves, 64x32 macro tile, k-chunked.
__global__ __launch_bounds__(256) void fi_kernel(const float* __restrict__ hw1,
                                                 const float* __restrict__ Su1,
                                                 const float* __restrict__ gb1,
                                                 float* __restrict__ fi) {
  __shared__ float Asm[64 * LDA];
  __shared__ float Bsm[32 * LDB];

  int tid = threadIdx.x;
  int wave = tid >> 5, lane = tid & 31;
  int mw = wave >> 1, nw = wave & 1;
  int row0_blk = blockIdx.y * 64;
  int n0_blk = blockIdx.x * 32;

  v8f acc = zero8();
  for (int kc = 0; kc < kH / KC; ++kc) {
    int k0 = kc * KC;
    __syncthreads();
    for (int i = tid; i < 64 * KC4; i += 256) {
      int r = i >> 3;                // KC4 == 8
      int c = (i & 7) << 2;
      *(float4*)&Asm[r * LDA + c] =
          *(const float4*)(hw1 + (size_t)(row0_blk + r) * kH + k0 + c);
    }
    for (int i = tid; i < 32 * KC4; i += 256) {
      int r = i >> 3;
      int c = (i & 7) << 2;
      *(float4*)&Bsm[r * LDB + c] =
          *(const float4*)(Su1 + (size_t)(n0_blk + r) * kH + k0 + c);
    }
    __syncthreads();
    const float* Ap = &Asm[(mw * 16 + (lane & 15)) * LDA + ((lane >> 4) << 1)];
    const float* Bp = &Bsm[(nw * 16 + (lane & 15)) * LDB + ((lane >> 4) << 1)];
#pragma unroll
    for (int k = 0; k < KC; k += 4) {
      v2f a; a.x = Ap[k]; a.y = Ap[k + 1];
      v2f b; b.x = Bp[k]; b.y = Bp[k + 1];
      acc = wmma4(a, b, acc);
    }
  }
  int row0 = row0_blk + mw * 16;
  int hcol = n0_blk + nw * 16 + (lane & 15);
#pragma unroll
  for (int e = 0; e < 8; ++e) {
    int m = e + ((lane >> 4) << 3);
    int row = row0 + m;
    int b = row % kB;
    fi[(size_t)row * kH + hcol] = sigm(acc[e] + gb1[(size_t)b * kH + hcol]);
  }
}

// Stage 5: global cell: softmax(fi over valid s), c_g, h_g -> output row S
__global__ __launch_bounds__(256) void global_cell_kernel(
    const float* __restrict__ fi, const float* __restrict__ cw1,
    const float* __restrict__ fg, const float* __restrict__ og,
    const float* __restrict__ c0, const int* __restrict__ seq_lens,
    float* __restrict__ out_h, float* __restrict__ out_c) {
  int idx = blockIdx.x * 256 + threadIdx.x;  // b*H + h
  if (idx >= kBH) return;
  int b = idx / kH;
  int len = seq_lens[b];
  float mx = -3.4e38f;
  for (int s = 0; s < len; ++s) mx = fmaxf(mx, fi[(size_t)s * kBH + idx]);
  float den = 0.f, num = 0.f;
  for (int s = 0; s < len; ++s) {
    float e = expf(fi[(size_t)s * kBH + idx] - mx);
    den += e;
    num += e * cw1[(size_t)s * kBH + idx];
  }
  float cg1 = c0[(size_t)kS * kBH + idx];
  float cg = fg[idx] * cg1 + num / den;
  float hg = og[idx] * tanhf(cg);
  out_h[(size_t)kS * kBH + idx] = hg;
  out_c[(size_t)kS * kBH + idx] = cg;
}

// Stage 6: fused 7-gate pre-activation GEMM (K = 3H window + D input) + gate softmax +
// c-window combine. 256 threads = 8 waves; block computes 64 rows x 32 cols; each wave
// holds its 16x16 tile for all 7 gates (7 accumulators). K staged through LDS in
// 32-wide chunks. The 7-gate weight panel (KC x 32 x 7, strides ldw / H*ldw) is moved
// by the Tensor Data Mover with hardware row-padding to the LDS pitch; the A panel is
// loaded manually with zero-fill (realizes the zero-padded +-1 window, no divergence
// around the WMMAs).
__global__ __launch_bounds__(256) void pre_gate_kernel(
    const float* __restrict__ hw1, const float* __restrict__ cw1,
    const float* __restrict__ seqs,
    const float* __restrict__ Ww, const float* __restrict__ Wu,
    const float* __restrict__ vterm,
    const float* __restrict__ c0, const int* __restrict__ seq_lens,
    float* __restrict__ out_h, float* __restrict__ out_c) {
  __shared__ float Asm[64 * LDA];          // 9.2 KB
  __shared__ float Bsm[7 * 32 * LDB];      // 32.3 KB

  int tid = threadIdx.x;
  int wave = tid >> 5, lane = tid & 31;
  int mw = wave >> 1, nw = wave & 1;
  int row0_blk = blockIdx.y * 64;
  int n0_blk = blockIdx.x * 32;

  v8f acc[7];
#pragma unroll
  for (int g = 0; g < 7; ++g) acc[g] = zero8();

  // t=0,1,2: h-window segments (row shift -B,0,+B ; Ww k-offsets 0,H,2H ; ld=3H)
  // t=3    : seqs vs Wu (ld=D)
  for (int t = 0; t < 4; ++t) {
    const float* Abase;
    int lda, R0, ldw, koff;
    const float* Wbase;
    if (t < 3) { Abase = hw1;  lda = kH; R0 = row0_blk + (t - 1) * kB; Wbase = Ww; ldw = 3 * kH; koff = t * kH; }
    else       { Abase = seqs; lda = kD; R0 = row0_blk;               Wbase = Wu; ldw = kD;     koff = 0; }

    for (int kc = 0; kc < kH / KC; ++kc) {
      int k0 = kc * KC;
      __syncthreads();   // previous chunk fully consumed before LDS is overwritten
      // A panel: 64 rows x KC, zero-filled outside [0, M)
      for (int i = tid; i < 64 * KC4; i += 256) {
        int r = i >> 3;
        int c = (i & 7) << 2;
        int grow = R0 + r;
        float4 v = {0.f, 0.f, 0.f, 0.f};
        if (grow >= 0 && grow < kM)
          v = *(const float4*)(Abase + (size_t)grow * lda + k0 + c);
        *(float4*)&Asm[r * LDA + c] = v;
      }
#if HAVE_TDM
      // B panels: one TDM descriptor moves the 3-D tile (KC x 32 cols x 7 gates) into
      // LDS, padding each KC-row (32 DWORDs) by 4 DWORDs -> LDS pitch LDB = 36 floats.
      if (wave == 0) {
        unsigned long long gaddr =
            (unsigned long long)(uintptr_t)Wbase +
            4ull * ((unsigned long long)n0_blk * ldw + (unsigned long long)(koff + k0));
        unsigned strideZ = (unsigned)kH * (unsigned)ldw;  // gate pitch, elements
        v4u g0;
        g0[0] = 1u;                                             // count=1 (valid D#)
        g0[1] = (unsigned)(uintptr_t)&Bsm[0];                   // lds_addr (bytes)
        g0[2] = (unsigned)gaddr;                                // global_addr[31:0]
        g0[3] = ((unsigned)(gaddr >> 32) & 0x01FFFFFFu) | (2u << 30);  // addr[56:32] | type=2
        v8i g1;
        g1[0] = (2 << 16)        // data_size = 4B
              | (1 << 20)        // pad_enable
              | (4 << 22)        // pad_interval: 32 DWORDs (one KC row)
              | (3 << 25);       // pad_amount: 4 DWORDs
        g1[1] = (KC & 0xFFFF) << 16;       // tensor_dim0[15:0] = KC
        g1[2] = (32 & 0xFFFF) << 16;       // tensor_dim0[31:16]=0 | tensor_dim1[15:0]=32
        g1[3] = (KC & 0xFFFF) << 16;       // tensor_dim1[31:16]=0 | tile_dim0 = KC
        g1[4] = 32 | (7 << 16);            // tile_dim1 = 32, tile_dim2 = 7
        g1[5] = ldw;                       // tensor_dim0_stride[31:0] (elements)
        g1[6] = (int)((strideZ & 0xFFFFu) << 16);  // dim0_stride hi=0 | dim1_stride[15:0]
        g1[7] = (int)(strideZ >> 16);              // dim1_stride[47:16]
        v4i g2;
        g2[0] = 7;  g2[1] = 0;  g2[2] = 0;  g2[3] = 0;   // tensor_dim2 = 7
        v4i g3;
        g3[0] = 0;  g3[1] = 0;  g3[2] = 0;  g3[3] = 0;
        v8i g4;                                           // extra operand (clang-23 6-arg
        g4[0] = 0; g4[1] = 0; g4[2] = 0; g4[3] = 0;       // form); unused state, zeroed
        g4[4] = 0; g4[5] = 0; g4[6] = 0; g4[7] = 0;
        __builtin_amdgcn_tensor_load_to_lds(g0, g1, g2, g3, g4, 0);
        __builtin_amdgcn_s_wait_tensorcnt(0);
      }
#else
      for (int i = tid; i < 7 * 32 * KC4; i += 256) {
        int rn = i >> 3;                 // g*32 + n
        int c = (i & 7) << 2;
        int g = rn >> 5, n = rn & 31;
        const float* src = Wbase + ((size_t)g * kH + n0_blk + n) * ldw + koff + k0 + c;
        *(float4*)&Bsm[rn * LDB + c] = *(const float4*)src;
      }
#endif
      __syncthreads();
      const float* Ap = &Asm[(mw * 16 + (lane & 15)) * LDA + ((lane >> 4) << 1)];
      const float* Bp = &Bsm[(nw * 16 + (lane & 15)) * LDB + ((lane >> 4) << 1)];
#pragma unroll
      for (int k = 0; k < KC; k += 4) {
        v2f a; a.x = Ap[k]; a.y = Ap[k + 1];
#pragma unroll
        for (int g = 0; g < 7; ++g) {
          const float* bp = Bp + g * 32 * LDB;
          v2f b; b.x = bp[k]; b.y = bp[k + 1];
          acc[g] = wmma4(a, b, acc[g]);
        }
      }
    }
  }

  int row0 = row0_blk + mw * 16;
  int n0 = n0_blk + nw * 16;
  int s = row0 / kB;  // uniform within a 16-row tile (B=32 multiple of 16)
  int hcol = n0 + (lane & 15);
#pragma unroll
  for (int e = 0; e < 8; ++e) {
    int m = e + ((lane >> 4) << 3);
    int row = row0 + m;
    int b = row % kB;
    int len = seq_lens[b];
    bool masked = (s >= len);
    float v[7];
#pragma unroll
    for (int g = 0; g < 7; ++g) v[g] = acc[g][e] + vterm[(size_t)g * kBH + (size_t)b * kH + hcol];
    // pre order: i, l, r, f, s, o, u
    float i_ = sigm(v[0]);
    float l_ = sigm(v[1]);
    float r_ = sigm(v[2]);
    float f_ = sigm(v[3]);
    float s_ = sigm(v[4]);
    float o_ = sigm(v[5]);
    float u_ = tanhf(v[6]);
    // softmax over gate axis, order (l, f, r, s, i)
    float mx = fmaxf(fmaxf(fmaxf(l_, f_), fmaxf(r_, s_)), i_);
    float e0 = expf(l_ - mx), e1 = expf(f_ - mx), e2 = expf(r_ - mx), e3 = expf(s_ - mx), e4 = expf(i_ - mx);
    float den = e0 + e1 + e2 + e3 + e4;
    size_t cidx = (size_t)row * kH + hcol;  // == s*B*H + b*H + h
    float cl = (s >= 1)     ? cw1[cidx - kBH] : 0.0f;
    float cc = cw1[cidx];
    float cr = (s + 1 < kS) ? cw1[cidx + kBH] : 0.0f;
    float cg1 = c0[(size_t)kS * kBH + (size_t)b * kH + hcol];
    float cw = (e0 * cl + e1 * cc + e2 * cr + e3 * cg1 + e4 * u_) / den;
    if (masked) cw = 0.0f;
    float hw = o_ * tanhf(cw);
    out_h[cidx] = hw;
    out_c[cidx] = cw;
  }
}

}  // namespace

extern "C" void kernel_launch(void* const* d_in, const int* in_sizes, int n_in,
                              void* d_out, int out_size, void* d_ws, size_t ws_size,
                              hipStream_t stream) {
  const float* seqs     = (const float*)d_in[0];
  const int*   seq_lens = (const int*)d_in[1];
  const float* h0       = (const float*)d_in[2];
  const float* c0       = (const float*)d_in[3];
  const float* Ww       = (const float*)d_in[4];
  const float* Wu       = (const float*)d_in[5];
  const float* Wv       = (const float*)d_in[6];
  const float* Wb       = (const float*)d_in[7];
  const float* Sw       = (const float*)d_in[8];
  const float* Su       = (const float*)d_in[9];
  const float* Sb       = (const float*)d_in[10];

  float* out_h = (float*)d_out;                       // (S+1, B, H)
  float* out_c = out_h + (size_t)(kS + 1) * kBH;      // (S+1, B, H)

  float* ws    = (float*)d_ws;
  float* hw1   = ws;               // S*B*H
  float* cw1   = hw1 + kSBH;       // S*B*H
  float* fi    = cw1 + kSBH;       // S*B*H
  float* hhat  = fi + kSBH;        // B*H
  float* fg    = hhat + kBH;       // B*H
  float* og    = fg + kBH;         // B*H
  float* gb1   = og + kBH;         // B*H
  float* vterm = gb1 + kBH;        // 7 * B*H  (includes Wb)

  prep_kernel<<<(kSBH + 255) / 256, 256, 0, stream>>>(h0, c0, seq_lens, hw1, cw1);
  hhat_kernel<<<(kBH + 255) / 256, 256, 0, stream>>>(hw1, hhat);
  small_gemm_kernel<<<dim3(kH / 16, kB / 16, 10), 32, 0, stream>>>(
      h0, hhat, Sw, Su, Sb, Wv, Wb, fg, og, gb1, vterm);
  fi_kernel<<<dim3(kH / 32, kM / 64), 256, 0, stream>>>(hw1, Su + (size_t)kH * kH, gb1, fi);
  global_cell_kernel<<<(kBH + 255) / 256, 256, 0, stream>>>(
      fi, cw1, fg, og, c0, seq_lens, out_h, out_c);
  pre_gate_kernel<<<dim3(kH / 32, kM / 64), 256, 0, stream>>>(
      hw1, cw1, seqs, Ww, Wu, vterm, c0, seq_lens, out_h, out_c);
}